// detector_21517786153156
// MI455X (gfx1250) — compile-verified
//
#include <hip/hip_runtime.h>
#include <math.h>
#include <stdint.h>

// Problem constants (from reference)
#define BB 32
#define RR 2000
#define CC 81
#define CM1 80          // C-1
#define NPAD 2048       // sort size (next pow2 >= 2000)
#define SORT_THREADS 256
#define EPT 8           // elements per thread (NPAD / SORT_THREADS)

// Bank-swizzle that permutes only the low 3 bits of a u64 slot index with a
// function of the 32-element window index. Bijective, keeps every element in
// its logical 8-block, and makes own-block accesses LDS-bank conflict free.
__device__ __forceinline__ int phys(int x) { return x ^ ((x >> 5) & 7); }

// ---------------------------------------------------------------------------
// Kernel 1: per-row softmax over 81 classes, max_scores, pred_labels.
// One wave32 per row; lanes cover classes {lane, lane+32, lane+64}.
// ---------------------------------------------------------------------------
__global__ void softmax81_kernel(const float* __restrict__ cls,
                                 float* __restrict__ dist,
                                 float* __restrict__ max_scores,
                                 float* __restrict__ pred_labels)
{
    const int wave = threadIdx.x >> 5;
    const int lane = threadIdx.x & 31;
    const int row  = blockIdx.x * 8 + wave;
    if (row >= BB * RR) return;

    const float* x = cls + (size_t)row * CC;
    const bool has2 = (lane + 64) < CC;   // lanes 0..16
    float v0 = x[lane];
    float v1 = x[lane + 32];
    float v2 = has2 ? x[lane + 64] : -3.4e38f;

    float m = fmaxf(fmaxf(v0, v1), v2);
    #pragma unroll
    for (int off = 16; off > 0; off >>= 1)
        m = fmaxf(m, __shfl_xor(m, off, 32));

    float e0 = expf(v0 - m);
    float e1 = expf(v1 - m);
    float e2 = has2 ? expf(v2 - m) : 0.0f;

    float s = e0 + e1 + e2;
    #pragma unroll
    for (int off = 16; off > 0; off >>= 1)
        s += __shfl_xor(s, off, 32);

    float d0 = e0 / s;
    float d1 = e1 / s;
    float d2 = e2 / s;

    dist[(size_t)row * CC + lane]      = d0;
    dist[(size_t)row * CC + lane + 32] = d1;
    if (has2) dist[(size_t)row * CC + lane + 64] = d2;

    // composite (value, -index) max => first-occurrence argmax like jnp.argmax
    unsigned long long k =
        ((unsigned long long)__float_as_uint(d0) << 32) |
        (unsigned long long)(0xFFFFFFFFu - (unsigned)lane);
    unsigned long long k1 =
        ((unsigned long long)__float_as_uint(d1) << 32) |
        (unsigned long long)(0xFFFFFFFFu - (unsigned)(lane + 32));
    if (k1 > k) k = k1;
    if (has2) {
        unsigned long long k2 =
            ((unsigned long long)__float_as_uint(d2) << 32) |
            (unsigned long long)(0xFFFFFFFFu - (unsigned)(lane + 64));
        if (k2 > k) k = k2;
    }
    #pragma unroll
    for (int off = 16; off > 0; off >>= 1) {
        unsigned long long o = __shfl_xor(k, off, 32);
        if (o > k) k = o;
    }
    if (lane == 0) {
        max_scores[row]  = __uint_as_float((unsigned)(k >> 32));
        pred_labels[row] = (float)(0xFFFFFFFFu - (unsigned)(k & 0xFFFFFFFFu));
    }
}

// ---------------------------------------------------------------------------
// Kernel 2: transpose cls_prob[b, r, 1..80] into contiguous key columns
// keysT[b][c-1][r] (stride NPAD) so the sort kernel can async-DMA them.
// key = (float_bits(score) if score>0.1 else 0) << 32  |  (0xFFFFFFFF - r)
// ---------------------------------------------------------------------------
__global__ void build_keys_kernel(const float* __restrict__ cls,
                                  unsigned long long* __restrict__ keysT)
{
    __shared__ float tile[32][33];
    const int b  = blockIdx.z;
    const int r0 = blockIdx.x * 32;
    const int c0 = blockIdx.y * 32;
    const int tx = threadIdx.x, ty = threadIdx.y;

    const int r = r0 + ty;
    const int c = c0 + tx + 1;            // classes 1..80
    float v = 0.0f;
    if (r < RR && c <= CM1) v = cls[((size_t)b * RR + r) * CC + c];
    tile[ty][tx] = v;
    __syncthreads();

    const int col = c0 + ty;              // c-1 index 0..79
    const int rr  = r0 + tx;
    if (col < CM1 && rr < RR) {
        const float sc = tile[tx][ty];
        const unsigned hi = (sc > 0.1f) ? __float_as_uint(sc) : 0u;
        keysT[(((size_t)b * CM1 + col) << 11) + rr] =
            ((unsigned long long)hi << 32) |
            (unsigned long long)(0xFFFFFFFFu - (unsigned)rr);
    }
}

// ---------------------------------------------------------------------------
// Kernel 3: per-(b, class) column. Async-DMA the 16KB key column into LDS,
// register-blocked bitonic sort (8 elems/thread: j<8 stages in VGPRs, j>=8
// in swizzled LDS), then decode + emit straight from registers.
// grid = (80, 32), block = 256.
// ---------------------------------------------------------------------------
__global__ void __launch_bounds__(SORT_THREADS)
sort_emit_kernel(const float* __restrict__ rois,
                 const float* __restrict__ cls,
                 const float* __restrict__ bbox,
                 const unsigned long long* __restrict__ keysT,
                 int use_ws,
                 float* __restrict__ dets,
                 float* __restrict__ sorted_valid)
{
    __shared__ unsigned long long keys[NPAD];
    const int cm1 = blockIdx.x;          // 0..79
    const int b   = blockIdx.y;          // 0..31
    const int c   = cm1 + 1;             // 1..80
    const int tid = threadIdx.x;
    const int i0  = tid << 3;            // this thread's logical block base

    unsigned long long reg[EPT];

    if (use_ws) {
        // CDNA5 async global->LDS DMA (b128 per lane), tracked by ASYNCcnt.
        const unsigned long long* src = keysT + (((size_t)b * CM1 + cm1) << 11);
        #pragma unroll
        for (int i = tid; i < NPAD / 2; i += SORT_THREADS) {
            unsigned lds_addr = (unsigned)(uintptr_t)(&keys[i * 2]);
            unsigned long long gaddr = (unsigned long long)(uintptr_t)(src + i * 2);
            asm volatile("global_load_async_to_lds_b128 %0, %1, off"
                         :: "v"(lds_addr), "v"(gaddr) : "memory");
        }
        asm volatile("s_wait_asynccnt 0" ::: "memory");
        __syncthreads();
        #pragma unroll
        for (int o = 0; o < EPT; ++o)
            reg[o] = (i0 + o < RR) ? keys[i0 + o] : 0ull;  // pads -> 0 in regs
    } else {
        // fallback: build keys from strided cls_prob loads (L2-resident)
        #pragma unroll
        for (int o = 0; o < EPT; ++o) {
            const int i = i0 + o;
            unsigned long long key = 0ull;
            if (i < RR) {
                const float sc = cls[((size_t)b * RR + i) * CC + c];
                const unsigned hi = (sc > 0.1f) ? __float_as_uint(sc) : 0u;
                key = ((unsigned long long)hi << 32) |
                      (unsigned long long)(0xFFFFFFFFu - (unsigned)i);
            }
            reg[o] = key;
        }
    }

    // In-register compare-exchange stage (j in {1,2,4}), descending network.
    auto reg_stage = [&](int j, int k) {
        #pragma unroll
        for (int o = 0; o < EPT; ++o) {
            const int oo = o ^ j;
            if (oo > o) {
                const bool down = (((i0 + o) & k) == 0);
                const unsigned long long a = reg[o], bv = reg[oo];
                if ((a < bv) == down) { reg[o] = bv; reg[oo] = a; }
            }
        }
    };

    // Phases k = 2, 4, 8: fully intra-thread, no LDS, no barriers.
    reg_stage(1, 2);
    reg_stage(2, 4); reg_stage(1, 4);
    reg_stage(4, 8); reg_stage(2, 8); reg_stage(1, 8);

    // Phases k = 16 .. 2048: j >= 8 via swizzled LDS, j = 4,2,1 in registers.
    for (int k = 16; k <= NPAD; k <<= 1) {
        #pragma unroll
        for (int o = 0; o < EPT; ++o) keys[phys(i0 + o)] = reg[o]; // own block
        __syncthreads();
        for (int j = k >> 1; j >= 8; j >>= 1) {
            if ((i0 & j) == 0) {     // owner of the low element of each pair
                #pragma unroll
                for (int o = 0; o < EPT; ++o) {
                    const int i = i0 + o;
                    const int l = i ^ j;
                    const unsigned long long a  = keys[phys(i)];
                    const unsigned long long bv = keys[phys(l)];
                    const bool down = ((i & k) == 0);
                    if ((a < bv) == down) { keys[phys(i)] = bv; keys[phys(l)] = a; }
                }
            }
            __syncthreads();
        }
        #pragma unroll
        for (int o = 0; o < EPT; ++o) reg[o] = keys[phys(i0 + o)];  // own block
        reg_stage(4, k); reg_stage(2, k); reg_stage(1, k);
    }

    // Emit sorted detections directly from registers (positions i0..i0+7).
    #pragma unroll
    for (int o = 0; o < EPT; ++o) {
        const int j = i0 + o;
        if (j >= RR) continue;
        const unsigned long long key = reg[o];
        const unsigned hi = (unsigned)(key >> 32);
        const size_t row      = (size_t)b * RR + j;
        const size_t det_base = (row * CM1 + cm1) * 5;
        const size_t sv_idx   =  row * CM1 + cm1;

        if (hi != 0u) {
            const int   r     = (int)(0xFFFFFFFFu - (unsigned)(key & 0xFFFFFFFFu));
            const float score = __uint_as_float(hi);

            const float* roi = rois + ((size_t)b * RR + r) * 5;
            const float x1 = roi[1], y1 = roi[2], x2 = roi[3], y2 = roi[4];
            const float w  = x2 - x1 + 1.0f;
            const float h  = y2 - y1 + 1.0f;
            const float cx = x1 + 0.5f * w;
            const float cy = y1 + 0.5f * h;

            const float* dp = bbox + ((size_t)b * RR + r) * (4 * CC) + 4 * c;
            const float dx = dp[0] * 0.1f;
            const float dy = dp[1] * 0.1f;
            const float dw = dp[2] * 0.2f;
            const float dh = dp[3] * 0.2f;

            const float pcx = dx * w + cx;
            const float pcy = dy * h + cy;
            const float pw  = expf(dw) * w;
            const float ph  = expf(dh) * h;

            dets[det_base + 0] = fmaxf(pcx - 0.5f * pw, 0.0f);
            dets[det_base + 1] = fmaxf(pcy - 0.5f * ph, 0.0f);
            dets[det_base + 2] = fmaxf(pcx + 0.5f * pw - 1.0f, 0.0f);
            dets[det_base + 3] = fmaxf(pcy + 0.5f * ph - 1.0f, 0.0f);
            dets[det_base + 4] = score;
            sorted_valid[sv_idx] = 1.0f;
        } else {
            dets[det_base + 0] = 0.0f;
            dets[det_base + 1] = 0.0f;
            dets[det_base + 2] = 0.0f;
            dets[det_base + 3] = 0.0f;
            dets[det_base + 4] = 0.0f;
            sorted_valid[sv_idx] = 0.0f;
        }
    }
}

// ---------------------------------------------------------------------------
extern "C" void kernel_launch(void* const* d_in, const int* in_sizes, int n_in,
                              void* d_out, int out_size, void* d_ws, size_t ws_size,
                              hipStream_t stream)
{
    const float* rois = (const float*)d_in[0];   // (32,2000,5)
    const float* cls  = (const float*)d_in[1];   // (32,2000,81)
    const float* bbox = (const float*)d_in[2];   // (32,2000,324)

    float* out          = (float*)d_out;
    float* dets         = out;                                   // 25,600,000
    float* sorted_valid = out + (size_t)BB * RR * CM1 * 5;       //  5,120,000
    float* dist         = sorted_valid + (size_t)BB * RR * CM1;  //  5,184,000
    float* max_scores   = dist + (size_t)BB * RR * CC;           //     64,000
    float* pred_labels  = max_scores + (size_t)BB * RR;          //     64,000

    softmax81_kernel<<<(BB * RR) / 8, 256, 0, stream>>>(cls, dist, max_scores,
                                                        pred_labels);

    const size_t need = (size_t)BB * CM1 * NPAD * sizeof(unsigned long long);
    const int use_ws = (ws_size >= need) ? 1 : 0;
    unsigned long long* keysT = (unsigned long long*)d_ws;
    if (use_ws) {
        dim3 g((RR + 31) / 32, (CM1 + 31) / 32, BB);
        dim3 blk(32, 32);
        build_keys_kernel<<<g, blk, 0, stream>>>(cls, keysT);
    }

    sort_emit_kernel<<<dim3(CM1, BB), SORT_THREADS, 0, stream>>>(
        rois, cls, bbox, keysT, use_ws, dets, sorted_valid);
}